// GATNet_33852932227160
// MI455X (gfx1250) — compile-verified
//
#include <hip/hip_runtime.h>
#include <hip/hip_bf16.h>

#define N_SRC  50000
#define N_DST1 20000
#define N_DST2 10000
#define E1     320000
#define E2     160000
#define TT     12
#define C_IN   64
#define HID    16
#define C_OUT  32

typedef __attribute__((ext_vector_type(2))) float v2f;
typedef __attribute__((ext_vector_type(8))) float v8f;

// ---------------------------------------------------------------- utilities
__global__ void fill_f32(float* __restrict__ p, int n, float v) {
    int i = blockIdx.x * blockDim.x + threadIdx.x;
    if (i < n) p[i] = v;
}

__global__ void leaky_inplace(float* __restrict__ p, int n, float slope) {
    int i = blockIdx.x * blockDim.x + threadIdx.x;
    if (i < n) { float x = p[i]; p[i] = x > 0.f ? x : slope * x; }
}

// ------------------------------------------------- layer 1 projection (WMMA)
// h[n][t][16] = x[t][n][0:64] @ W1[64][16], fp32 WMMA 16x16x4, K=64 (16 steps)
__global__ void proj1_wmma(const float* __restrict__ x, const float* __restrict__ W,
                           float* __restrict__ h) {
    const int nTiles = N_SRC / 16;                       // 3125 (exact)
    int wave = (blockIdx.x * blockDim.x + threadIdx.x) >> 5;
    int lane = threadIdx.x & 31;
    if (wave >= nTiles * TT) return;                     // wave-uniform exit
    int tile = wave % nTiles;
    int t    = wave / nTiles;
    int n0   = tile * 16;
    int m    = lane & 15;                                // A-row / B-col / D-col
    int kh   = (lane >> 4) * 2;                          // K sub-offset 0 or 2
    const float* xrow = x + ((size_t)t * N_SRC + (n0 + m)) * C_IN;

    v8f c = {};
    #pragma unroll
    for (int k0 = 0; k0 < C_IN; k0 += 4) {
        v2f a, b;
        a.x = xrow[k0 + kh];
        a.y = xrow[k0 + kh + 1];
        b.x = W[(k0 + kh)     * HID + m];
        b.y = W[(k0 + kh + 1) * HID + m];
        c = __builtin_amdgcn_wmma_f32_16x16x4_f32(false, a, false, b,
                                                  (short)0, c, false, false);
    }
    int o = lane & 15;
    int mbase = (lane >> 4) * 8;
    #pragma unroll
    for (int i = 0; i < 8; ++i) {
        int n = n0 + mbase + i;
        h[((size_t)n * TT + t) * HID + o] = c[i];
    }
}

// ------------------------------------------------- layer 2 projection (WMMA)
// h2p[n][t][32] = h1[n][t][0:16] @ W2[16][32], K=16 (4 steps), two N-halves
__global__ void proj2_wmma(const float* __restrict__ h1, const float* __restrict__ W,
                           float* __restrict__ h2p) {
    const int nTiles = N_DST1 / 16;                      // 1250 (exact)
    int wave = (blockIdx.x * blockDim.x + threadIdx.x) >> 5;
    int lane = threadIdx.x & 31;
    if (wave >= nTiles * TT) return;
    int tile = wave % nTiles;
    int t    = wave / nTiles;
    int n0   = tile * 16;
    int m    = lane & 15;
    int kh   = (lane >> 4) * 2;
    const float* row = h1 + ((size_t)(n0 + m) * TT + t) * HID;

    v2f a[4];
    #pragma unroll
    for (int kc = 0; kc < 4; ++kc) {
        a[kc].x = row[kc * 4 + kh];
        a[kc].y = row[kc * 4 + kh + 1];
    }
    int o = lane & 15;
    int mbase = (lane >> 4) * 8;
    #pragma unroll
    for (int half = 0; half < 2; ++half) {
        v8f c = {};
        #pragma unroll
        for (int kc = 0; kc < 4; ++kc) {
            int k0 = kc * 4;
            v2f b;
            b.x = W[(k0 + kh)     * C_OUT + half * 16 + m];
            b.y = W[(k0 + kh + 1) * C_OUT + half * 16 + m];
            c = __builtin_amdgcn_wmma_f32_16x16x4_f32(false, a[kc], false, b,
                                                      (short)0, c, false, false);
        }
        #pragma unroll
        for (int i = 0; i < 8; ++i) {
            int n = n0 + mbase + i;
            h2p[((size_t)n * TT + t) * C_OUT + half * 16 + o] = c[i];
        }
    }
}

// --------------------------------------------- per-node attention dot terms
__global__ void elre_kernel(const float* __restrict__ h, const float* __restrict__ al,
                            const float* __restrict__ ar, float* __restrict__ el,
                            float* __restrict__ er, int N, int Ndst, int O) {
    int i = blockIdx.x * blockDim.x + threadIdx.x;       // i = n*T + t
    if (i >= N * TT) return;
    const float* p = h + (size_t)i * O;
    float sl = 0.f, sr = 0.f;
    for (int o = 0; o < O; ++o) { float v = p[o]; sl += v * al[o]; sr += v * ar[o]; }
    el[i] = sl;
    if (i < Ndst * TT) er[i] = sr;
}

// ------------------------------------------------------ segment softmax pass1
__device__ __forceinline__ unsigned flip_f32(float f) {
    unsigned u = __float_as_uint(f);
    return (u & 0x80000000u) ? ~u : (u | 0x80000000u);
}
__device__ __forceinline__ float unflip_f32(unsigned u) {
    return (u & 0x80000000u) ? __uint_as_float(u ^ 0x80000000u) : __uint_as_float(~u);
}

__global__ void edge_logit_max(const int* __restrict__ src, const int* __restrict__ dst,
                               const float* __restrict__ el, const float* __restrict__ er,
                               float* __restrict__ e_out, unsigned* __restrict__ mmax, int E) {
    int i = blockIdx.x * blockDim.x + threadIdx.x;       // i = edge*T + t
    if (i >= E * TT) return;
    int edge = i / TT, t = i % TT;
    float v = el[src[edge] * TT + t] + er[dst[edge] * TT + t];
    v = v > 0.f ? v : 0.2f * v;                          // leaky_relu slope 0.2
    e_out[i] = v;
    atomicMax(&mmax[dst[edge] * TT + t], flip_f32(v));
}

// ------------------------------------------------------ segment softmax pass2
__global__ void edge_expsum(const int* __restrict__ dst, float* __restrict__ e,
                            const unsigned* __restrict__ mmax, float* __restrict__ s, int E) {
    int i = blockIdx.x * blockDim.x + threadIdx.x;
    if (i >= E * TT) return;
    int edge = i / TT, t = i % TT;
    int d = dst[edge];
    float ex = __expf(e[i] - unflip_f32(mmax[d * TT + t]));
    e[i] = ex;
    atomicAdd(&s[d * TT + t], ex);
}

// ------------------------------------------------------ message accumulation
__global__ void edge_acc16(const int* __restrict__ src, const int* __restrict__ dst,
                           const float* __restrict__ ew, const float* __restrict__ ex,
                           const float* __restrict__ s, const float* __restrict__ hsrc,
                           float* __restrict__ out, int E) {
    int i = blockIdx.x * blockDim.x + threadIdx.x;
    if (i >= E * TT) return;
    int edge = i / TT, t = i % TT;
    int d = dst[edge];
    float coef = ex[i] / (s[d * TT + t] + 1e-9f) * ew[edge];
    const float4* hp = (const float4*)(hsrc + ((size_t)src[edge] * TT + t) * HID);
    float* op = out + ((size_t)d * TT + t) * HID;
    #pragma unroll
    for (int q = 0; q < 4; ++q) {
        float4 v = hp[q];
        atomicAdd(&op[q * 4 + 0], coef * v.x);
        atomicAdd(&op[q * 4 + 1], coef * v.y);
        atomicAdd(&op[q * 4 + 2], coef * v.z);
        atomicAdd(&op[q * 4 + 3], coef * v.w);
    }
}

// layer 2 accumulates directly into transposed output: out[(t*N_DST2+d)*32+o]
__global__ void edge_acc32_t(const int* __restrict__ src, const int* __restrict__ dst,
                             const float* __restrict__ ew, const float* __restrict__ ex,
                             const float* __restrict__ s, const float* __restrict__ hsrc,
                             float* __restrict__ out, int E) {
    int i = blockIdx.x * blockDim.x + threadIdx.x;
    if (i >= E * TT) return;
    int edge = i / TT, t = i % TT;
    int d = dst[edge];
    float coef = ex[i] / (s[d * TT + t] + 1e-9f) * ew[edge];
    const float4* hp = (const float4*)(hsrc + ((size_t)src[edge] * TT + t) * C_OUT);
    float* op = out + ((size_t)t * N_DST2 + d) * C_OUT;
    #pragma unroll
    for (int q = 0; q < 8; ++q) {
        float4 v = hp[q];
        atomicAdd(&op[q * 4 + 0], coef * v.x);
        atomicAdd(&op[q * 4 + 1], coef * v.y);
        atomicAdd(&op[q * 4 + 2], coef * v.z);
        atomicAdd(&op[q * 4 + 3], coef * v.w);
    }
}

// ---------------------------------------------------------------- launcher
static inline int cdiv(long long a, int b) { return (int)((a + b - 1) / b); }

extern "C" void kernel_launch(void* const* d_in, const int* in_sizes, int n_in,
                              void* d_out, int out_size, void* d_ws, size_t ws_size,
                              hipStream_t stream) {
    const float* node_feat = (const float*)d_in[0];
    const int*   esrc1 = (const int*)d_in[1];
    const int*   edst1 = (const int*)d_in[2];
    const float* ew1   = (const float*)d_in[3];
    const int*   esrc2 = (const int*)d_in[4];
    const int*   edst2 = (const int*)d_in[5];
    const float* ew2   = (const float*)d_in[6];
    const float* W1  = (const float*)d_in[7];
    const float* al1 = (const float*)d_in[8];
    const float* ar1 = (const float*)d_in[9];
    const float* W2  = (const float*)d_in[10];
    const float* al2 = (const float*)d_in[11];
    const float* ar2 = (const float*)d_in[12];
    float* out = (float*)d_out;
    float* ws  = (float*)d_ws;

    // workspace carving (floats); regions reused across layers
    float* hsrc1 = ws;                                   // 9,600,000  (also h2p)
    float* ebuf  = ws + 9600000;                         // 3,840,000  (e1 / e2)
    float* el    = ws + 13440000;                        //   600,000  (el1 / el2)
    float* er    = ws + 14040000;                        //   240,000  (er1 / er2)
    float* mbuf  = ws + 14280000;                        //   240,000  (flipped-uint max)
    float* sbuf  = ws + 14520000;                        //   240,000  (exp sums)
    float* h1    = ws + 14760000;                        // 3,840,000
    float* h2p   = hsrc1;                                // alias (7,680,000 used)

    const int B = 256;

    // ---- layer 1 init
    fill_f32<<<cdiv(N_DST1 * TT * HID, B), B, 0, stream>>>(h1, N_DST1 * TT * HID, 0.f);
    fill_f32<<<cdiv(N_DST1 * TT, B), B, 0, stream>>>(mbuf, N_DST1 * TT, 0.f);
    fill_f32<<<cdiv(N_DST1 * TT, B), B, 0, stream>>>(sbuf, N_DST1 * TT, 0.f);

    // ---- layer 1 projection + attention terms
    {
        int waves = (N_SRC / 16) * TT;
        proj1_wmma<<<cdiv(waves, B / 32), B, 0, stream>>>(node_feat, W1, hsrc1);
    }
    elre_kernel<<<cdiv(N_SRC * TT, B), B, 0, stream>>>(hsrc1, al1, ar1, el, er,
                                                       N_SRC, N_DST1, HID);

    // ---- layer 1 segment softmax + aggregation
    edge_logit_max<<<cdiv((long long)E1 * TT, B), B, 0, stream>>>(
        esrc1, edst1, el, er, ebuf, (unsigned*)mbuf, E1);
    edge_expsum<<<cdiv((long long)E1 * TT, B), B, 0, stream>>>(
        edst1, ebuf, (const unsigned*)mbuf, sbuf, E1);
    edge_acc16<<<cdiv((long long)E1 * TT, B), B, 0, stream>>>(
        esrc1, edst1, ew1, ebuf, sbuf, hsrc1, h1, E1);
    leaky_inplace<<<cdiv(N_DST1 * TT * HID, B), B, 0, stream>>>(h1, N_DST1 * TT * HID, 0.01f);

    // ---- layer 2 init
    fill_f32<<<cdiv(N_DST2 * TT, B), B, 0, stream>>>(mbuf, N_DST2 * TT, 0.f);
    fill_f32<<<cdiv(N_DST2 * TT, B), B, 0, stream>>>(sbuf, N_DST2 * TT, 0.f);
    fill_f32<<<cdiv(TT * N_DST2 * C_OUT, B), B, 0, stream>>>(out, TT * N_DST2 * C_OUT, 0.f);

    // ---- layer 2 projection + attention terms
    {
        int waves = (N_DST1 / 16) * TT;
        proj2_wmma<<<cdiv(waves, B / 32), B, 0, stream>>>(h1, W2, h2p);
    }
    elre_kernel<<<cdiv(N_DST1 * TT, B), B, 0, stream>>>(h2p, al2, ar2, el, er,
                                                        N_DST1, N_DST2, C_OUT);

    // ---- layer 2 segment softmax + aggregation (into transposed d_out)
    edge_logit_max<<<cdiv((long long)E2 * TT, B), B, 0, stream>>>(
        esrc2, edst2, el, er, ebuf, (unsigned*)mbuf, E2);
    edge_expsum<<<cdiv((long long)E2 * TT, B), B, 0, stream>>>(
        edst2, ebuf, (const unsigned*)mbuf, sbuf, E2);
    edge_acc32_t<<<cdiv((long long)E2 * TT, B), B, 0, stream>>>(
        esrc2, edst2, ew2, ebuf, sbuf, h2p, out, E2);
    leaky_inplace<<<cdiv(TT * N_DST2 * C_OUT, B), B, 0, stream>>>(out, TT * N_DST2 * C_OUT, 0.01f);
}